// HardLinearAttention_14508399526246
// MI455X (gfx1250) — compile-verified
//
#include <hip/hip_runtime.h>

#define DD 2049   // 2d+1
#define DHALF 1024
#define NN 8192
#define N1 8193   // n+1

typedef float v2f __attribute__((ext_vector_type(2)));
typedef float v4f __attribute__((ext_vector_type(4)));
typedef float v8f __attribute__((ext_vector_type(8)));

// ---------------------------------------------------------------------------
// Kernel 1: bulk copy Z -> out (134 MB streaming). 4 independent b128 NT
// load/store pairs per loop iteration so loads clause together and stay in
// flight; NT hints keep the one-shot traffic out of L2.
// ---------------------------------------------------------------------------
__global__ void hla_copy_kernel(const float* __restrict__ Z,
                                float* __restrict__ out,
                                long long nvec, long long total) {
  const long long gid = (long long)blockIdx.x * blockDim.x + threadIdx.x;
  const long long S = (long long)gridDim.x * blockDim.x;
  const v4f* src = (const v4f*)Z;
  v4f* dst = (v4f*)out;
  long long i = gid;
  for (; i + 3 * S < nvec; i += 4 * S) {
    v4f t0 = __builtin_nontemporal_load(&src[i]);
    v4f t1 = __builtin_nontemporal_load(&src[i + S]);
    v4f t2 = __builtin_nontemporal_load(&src[i + 2 * S]);
    v4f t3 = __builtin_nontemporal_load(&src[i + 3 * S]);
    __builtin_nontemporal_store(t0, &dst[i]);
    __builtin_nontemporal_store(t1, &dst[i + S]);
    __builtin_nontemporal_store(t2, &dst[i + 2 * S]);
    __builtin_nontemporal_store(t3, &dst[i + 3 * S]);
  }
  for (; i < nvec; i += S) {
    v4f t = __builtin_nontemporal_load(&src[i]);
    __builtin_nontemporal_store(t, &dst[i]);
  }
  for (long long q = nvec * 4 + gid; q < total; q += S) out[q] = Z[q];
}

// ---------------------------------------------------------------------------
// Kernel 2: partial v[k] = sum_{j in split} Z[2d, j] * Z[k, j].
// grid = (64 k-blocks, 8 j-splits) -> 4096 waves. Each wave covers 128 j
// (32 WMMAs), batching 4 load-pairs per group so 8 b64 loads are in flight
// before the first v_wmma consumes them.
// ---------------------------------------------------------------------------
__global__ void hla_vdot_kernel(const float* __restrict__ Z,
                                float* __restrict__ pv) {
  const int tid  = threadIdx.x;
  const int lane = tid & 31;
  const int wave = tid >> 5;
  const int k0   = blockIdx.x * 16;
  const int js   = blockIdx.y;          // j-split 0..7

  const int m    = lane & 15;           // A: M row within the 16-row block
  const int koff = (lane >> 4) * 2;     // A: K=0,1 (lanes 0-15) / K=2,3 (16-31)

  const float* __restrict__ zrow = Z + (long long)(DD - 1) * N1;
  const float* __restrict__ arow = Z + (long long)(k0 + m) * N1;
  const float bmask = (m == 0) ? 1.0f : 0.0f;   // B column N=0 only

  v8f c = {};
  const int jbeg = js * 1024 + wave * 128;
  for (int j0 = jbeg; j0 < jbeg + 128; j0 += 16) {
    v2f a[4], b[4];
#pragma unroll
    for (int t = 0; t < 4; ++t) {
      const int jj = j0 + 4 * t + koff;
      a[t].x = __builtin_nontemporal_load(arow + jj);
      a[t].y = __builtin_nontemporal_load(arow + jj + 1);
      b[t].x = zrow[jj] * bmask;
      b[t].y = zrow[jj + 1] * bmask;
    }
#pragma unroll
    for (int t = 0; t < 4; ++t) {
      c = __builtin_amdgcn_wmma_f32_16x16x4_f32(
          false, a[t], false, b[t], (short)0, c, false, false);
    }
  }

  __shared__ float partial[8 * 16];
  if (m == 0) {                        // lanes 0 (M=0..7) and 16 (M=8..15)
    const int base = (lane >> 4) * 8;
#pragma unroll
    for (int i = 0; i < 8; ++i) partial[wave * 16 + base + i] = c[i];
  }
  __syncthreads();

  if (tid < 16) {
    float s = 0.0f;
#pragma unroll
    for (int w = 0; w < 8; ++w) s += partial[w * 16 + tid];
    pv[(long long)js * DHALF + k0 + tid] = s;
  }
}

// Combine the 8 j-split partials: v[k] = sum_js pv[js][k].
__global__ void hla_vreduce_kernel(const float* __restrict__ pv,
                                   float* __restrict__ v) {
  const int k = blockIdx.x * blockDim.x + threadIdx.x;
  if (k < DHALF) {
    float s = 0.0f;
#pragma unroll
    for (int w = 0; w < 8; ++w) s += pv[(long long)w * DHALF + k];
    v[k] = s;
  }
}

// ---------------------------------------------------------------------------
// Kernel 3: out[2d, j] = Z[2d, j] + (alpha/n) * sum_k v[k]*(Z[d+k,j]-Z[k,j]).
// 512-thread blocks: 32 j-columns x 16 k-chunks (64 k each) -> 4112 waves.
// v staged into LDS via async-to-LDS; 8 zt/zb pairs batched per group so 16
// loads are in flight per wave; deterministic LDS tree reduction.
// ---------------------------------------------------------------------------
__global__ void hla_wupdate_kernel(const float* __restrict__ Z,
                                   const float* __restrict__ v,
                                   const float* __restrict__ alpha,
                                   float* __restrict__ out) {
  __shared__ float vs[DHALF];
  __shared__ float red[16][32];
  const int tid = threadIdx.x;

  // Async copy v[0..1023] into LDS: first 256 threads x 16B = 4 KB.
  if (tid < 256) {
    unsigned lds_off = (unsigned)(size_t)(&vs[tid * 4]);
    unsigned long long gaddr = (unsigned long long)(size_t)(v + tid * 4);
    asm volatile("global_load_async_to_lds_b128 %0, %1, off"
                 :: "v"(lds_off), "v"(gaddr) : "memory");
  }
  asm volatile("s_wait_asynccnt 0" ::: "memory");
  __syncthreads();

  const int jlocal = tid & 31;
  const int kchunk = tid >> 5;                 // 0..15, 64 k's each
  const int kbase  = kchunk * 64;
  const long long j = (long long)blockIdx.x * 32 + jlocal;

  float acc = 0.0f;
  if (j < N1) {
    for (int kk = 0; kk < 64; kk += 8) {
      float zt[8], zb[8];
#pragma unroll
      for (int t = 0; t < 8; ++t) {
        const long long k = kbase + kk + t;
        zt[t] = __builtin_nontemporal_load(Z + k * N1 + j);
        zb[t] = __builtin_nontemporal_load(Z + (DHALF + k) * N1 + j);
      }
#pragma unroll
      for (int t = 0; t < 8; ++t) {
        acc += vs[kbase + kk + t] * (zb[t] - zt[t]);
      }
    }
  }
  red[kchunk][jlocal] = acc;
  __syncthreads();

  if (kchunk == 0 && j < N1) {
    float s = 0.0f;
#pragma unroll
    for (int w = 0; w < 16; ++w) s += red[w][jlocal];
    const float scale = alpha[0] * (1.0f / (float)NN);
    const long long idx = (long long)(DD - 1) * N1 + j;
    out[idx] = Z[idx] + scale * s;
  }
}

// ---------------------------------------------------------------------------
extern "C" void kernel_launch(void* const* d_in, const int* in_sizes, int n_in,
                              void* d_out, int out_size, void* d_ws, size_t ws_size,
                              hipStream_t stream) {
  (void)in_sizes; (void)n_in; (void)out_size; (void)ws_size;
  const float* Z     = (const float*)d_in[0];
  const float* alpha = (const float*)d_in[1];
  float* out = (float*)d_out;
  float* pv  = (float*)d_ws;                 // 8 x 1024 partial sums (32 KB)
  float* v   = pv + 8 * DHALF;               // final v (4 KB)

  const long long total = (long long)DD * N1;      // 16,787,457 floats
  const long long nvec  = total / 4;               // float4 vectors (+1 tail)

  hla_copy_kernel<<<4096, 256, 0, stream>>>(Z, out, nvec, total);
  dim3 vgrid(64, 8);
  hla_vdot_kernel<<<vgrid, 256, 0, stream>>>(Z, pv);
  hla_vreduce_kernel<<<4, 256, 0, stream>>>(pv, v);
  hla_wupdate_kernel<<<(N1 + 31) / 32, 512, 0, stream>>>(Z, v, alpha, out);
}